// VerifyReLU_38482906972594
// MI455X (gfx1250) — compile-verified
//
#include <hip/hip_runtime.h>

// DeepPoly ReLU relaxation — pure streaming elementwise kernel.
// Memory-bound: 320 MiB traffic, ~14.4 us floor at 23.3 TB/s on MI455X.
// Strategy: b128 non-temporal loads/stores (single-touch stream vs 192MB L2),
// wave32 blocks of 256, one float4 per lane, grid-stride + scalar tail.

typedef __attribute__((ext_vector_type(4))) float v4f;

struct Bounds { float lb, ub; };

__device__ __forceinline__ Bounds relu_relax(float l, float u, float a) {
    float ca    = fminf(fmaxf(a, 0.0f), 1.0f);       // clamp(alpha, 0, 1)
    float slope = fmaxf(u / (u - l), 0.0f);          // clamp(min=0)
    bool  pos   = l > 0.0f;
    float uc_bias = pos ? 0.0f : -(slope * l);
    float uc_diag = pos ? 1.0f : slope;
    float lc_diag = (u < 0.0f) ? 0.0f : ca;
    Bounds r;
    r.lb = l * lc_diag;
    r.ub = u * uc_diag + uc_bias;
    return r;
}

__global__ __launch_bounds__(256) void deeppoly_relu_kernel(
    const float* __restrict__ lb, const float* __restrict__ ub,
    const float* __restrict__ alpha, float* __restrict__ out_lb,
    float* __restrict__ out_ub, int n) {
    const int tid    = blockIdx.x * blockDim.x + threadIdx.x;
    const int stride = gridDim.x * blockDim.x;
    const int n4     = n >> 2;

    const v4f* __restrict__ lb4  = (const v4f*)lb;
    const v4f* __restrict__ ub4  = (const v4f*)ub;
    const v4f* __restrict__ al4  = (const v4f*)alpha;
    v4f* __restrict__ olb4 = (v4f*)out_lb;
    v4f* __restrict__ oub4 = (v4f*)out_ub;

    // Main body: one float4 per lane per iteration, non-temporal (single-touch).
    for (int i = tid; i < n4; i += stride) {
        v4f l = __builtin_nontemporal_load(&lb4[i]);
        v4f u = __builtin_nontemporal_load(&ub4[i]);
        v4f a = __builtin_nontemporal_load(&al4[i]);
        v4f olb, oub;
        Bounds r0 = relu_relax(l.x, u.x, a.x);
        Bounds r1 = relu_relax(l.y, u.y, a.y);
        Bounds r2 = relu_relax(l.z, u.z, a.z);
        Bounds r3 = relu_relax(l.w, u.w, a.w);
        olb.x = r0.lb; oub.x = r0.ub;
        olb.y = r1.lb; oub.y = r1.ub;
        olb.z = r2.lb; oub.z = r2.ub;
        olb.w = r3.lb; oub.w = r3.ub;
        __builtin_nontemporal_store(olb, &olb4[i]);
        __builtin_nontemporal_store(oub, &oub4[i]);
    }

    // Scalar tail (n not multiple of 4; not hit for N = 16M but keeps it general).
    for (int i = (n4 << 2) + tid; i < n; i += stride) {
        Bounds r = relu_relax(lb[i], ub[i], alpha[i]);
        __builtin_nontemporal_store(r.lb, &out_lb[i]);
        __builtin_nontemporal_store(r.ub, &out_ub[i]);
    }
}

extern "C" void kernel_launch(void* const* d_in, const int* in_sizes, int n_in,
                              void* d_out, int out_size, void* d_ws, size_t ws_size,
                              hipStream_t stream) {
    (void)n_in; (void)out_size; (void)d_ws; (void)ws_size;
    const float* lb    = (const float*)d_in[0];
    const float* ub    = (const float*)d_in[1];
    const float* alpha = (const float*)d_in[2];
    const int n = in_sizes[0];

    float* out_lb = (float*)d_out;       // first N elements
    float* out_ub = (float*)d_out + n;   // second N elements

    const int threads = 256;             // 8 wave32 waves per block
    int n4 = (n + 3) >> 2;
    int blocks = (n4 + threads - 1) / threads;
    if (blocks < 1) blocks = 1;
    if (blocks > (1 << 20)) blocks = (1 << 20);

    deeppoly_relu_kernel<<<blocks, threads, 0, stream>>>(lb, ub, alpha,
                                                         out_lb, out_ub, n);
}